// KVCache_84928683311337
// MI455X (gfx1250) — compile-verified
//
#include <hip/hip_runtime.h>

// KV-cache update + rotation: pure streaming gather/copy.
// B=8, S_NEW=1024, H=16, D=128, MAX_SIZE=8192 (shapes from reference).
#define B_     8
#define H_     16
#define D_     128
#define ROW_ELEMS (H_ * D_)       // 2048 fp16 per (b, s) row
#define ROW_VEC   (ROW_ELEMS / 8) // 256 x 16B chunks per row
#define ROWS_PER_BLOCK 4

// Native clang vector (not HIP_vector_type) so nontemporal builtins accept it.
typedef unsigned int v4u __attribute__((ext_vector_type(4)));

// ---- Kernel 1: init inverse-scatter map to -1 ------------------------------
__global__ void kv_map_init(int* __restrict__ map, int max_size) {
    int i = blockIdx.x * blockDim.x + threadIdx.x;
    if (i < max_size) map[i] = -1;
}

// ---- Kernel 2: map[pos[i] % max_size] = i ----------------------------------
__global__ void kv_map_scatter(const int* __restrict__ pos, int* __restrict__ map,
                               int s_new, int max_size) {
    int i = blockIdx.x * blockDim.x + threadIdx.x;
    if (i < s_new) {
        int p = pos[i] % max_size;   // reference wraps with % max_size
        if (p < 0) p += max_size;
        map[p] = i;
    }
}

// ---- Kernel 3: rotated gather of updated cache into d_out ------------------
// Block = 256 threads = one full (b,s) row (256 x 16B = 4KiB) per tensor.
// grid.x covers S_out/ROWS_PER_BLOCK rows, grid.y = batch.
// Bulk traffic: 128-bit ops with NT hints (working set >> 192MB L2).
__global__ __launch_bounds__(256) void kv_gather(
    const v4u* __restrict__ k,  const v4u* __restrict__ v,
    const v4u* __restrict__ kc, const v4u* __restrict__ vc,
    const int* __restrict__ map, const int* __restrict__ max_pos,
    v4u* __restrict__ out, int s_out_total, int s_new, int max_size) {

    const int inner = threadIdx.x;     // 16B chunk index within a row: 0..255
    const int b     = blockIdx.y;

    // Wave-uniform scalars -> scalar loads / SALU math.
    const int next  = *max_pos + 1;
    const int split = (next > max_size) ? (next % max_size) : 0;

    const size_t out_v_base = (size_t)B_ * (size_t)s_out_total * ROW_VEC;

#pragma unroll
    for (int r = 0; r < ROWS_PER_BLOCK; ++r) {
        int s = blockIdx.x * ROWS_PER_BLOCK + r;
        if (s < s_out_total) {
            int src = split + s;
            if (src >= max_size) src -= max_size;

            const int newi = map[src];          // uniform per row -> scalar load

            size_t soff;
            const v4u *ks, *vs;
            if (newi >= 0) {                    // row was overwritten by k/v
                soff = ((size_t)b * (size_t)s_new + (size_t)newi) * ROW_VEC + inner;
                ks = k  + soff;  vs = v  + soff;
            } else {                            // row comes from the old cache
                soff = ((size_t)b * (size_t)max_size + (size_t)src) * ROW_VEC + inner;
                ks = kc + soff;  vs = vc + soff;
            }

            const size_t o = ((size_t)b * (size_t)s_out_total + (size_t)s) * ROW_VEC + inner;

            v4u kd = __builtin_nontemporal_load(ks);   // global_load_b128 th:NT
            v4u vd = __builtin_nontemporal_load(vs);
            __builtin_nontemporal_store(kd, out + o);              // k_out half
            __builtin_nontemporal_store(vd, out + out_v_base + o); // v_out half
        }
    }
}

extern "C" void kernel_launch(void* const* d_in, const int* in_sizes, int n_in,
                              void* d_out, int out_size, void* d_ws, size_t ws_size,
                              hipStream_t stream) {
    const v4u* k  = (const v4u*)d_in[0];
    const v4u* v  = (const v4u*)d_in[1];
    const v4u* kc = (const v4u*)d_in[2];
    const v4u* vc = (const v4u*)d_in[3];
    const int* pos     = (const int*)d_in[4];
    const int* max_pos = (const int*)d_in[5];

    const int s_new    = in_sizes[0] / (B_ * ROW_ELEMS);   // 1024
    const int max_size = in_sizes[2] / (B_ * ROW_ELEMS);   // 8192
    const int s_out    = out_size / (2 * B_ * ROW_ELEMS);  // 8192 (rotated full window)

    int* map = (int*)d_ws;                                 // 32 KiB scratch

    kv_map_init<<<(max_size + 255) / 256, 256, 0, stream>>>(map, max_size);
    kv_map_scatter<<<(s_new + 255) / 256, 256, 0, stream>>>(pos, map, s_new, max_size);

    dim3 grid((s_out + ROWS_PER_BLOCK - 1) / ROWS_PER_BLOCK, B_);
    kv_gather<<<grid, 256, 0, stream>>>(k, v, kc, vc, map, max_pos,
                                        (v4u*)d_out, s_out, s_new, max_size);
}